// NCGNN_75402445848807
// MI455X (gfx1250) — compile-verified
//
#include <hip/hip_runtime.h>

typedef float v2f __attribute__((ext_vector_type(2)));
typedef float v8f __attribute__((ext_vector_type(8)));

#define BB 4
#define NN 256
#define DD 128
#define HD 256
#define MAXE 4096
#define BN_EPS 1e-5f

__device__ __forceinline__ v8f wmma_step(v2f a, v2f b, v8f c) {
  // V_WMMA_F32_16X16X4_F32 : D = A(16x4,f32) x B(4x16,f32) + C(16x16,f32)
  return __builtin_amdgcn_wmma_f32_16x16x4_f32(false, a, false, b, (short)0, c,
                                               false, false);
}

// CDNA5 async global->LDS copy (16B per lane), tracked by ASYNCcnt.
__device__ __forceinline__ void async_copy_b128(unsigned lds_off,
                                                const void* gaddr) {
  asm volatile("global_load_async_to_lds_b128 %0, %1, off"
               :
               : "v"(lds_off), "v"((unsigned long long)(uintptr_t)gaddr)
               : "memory");
}
__device__ __forceinline__ void wait_async() {
  asm volatile("s_wait_asynccnt 0x0" ::: "memory");
}

// ---------------------------------------------------------------------------
// K1: deterministic edge compaction. One block per batch, thread t owns row t.
// Upper-triangle edges (j<k) listed in row-major order via block-wide scan.
// ---------------------------------------------------------------------------
__global__ void edge_build(const float* __restrict__ A, int* __restrict__ cnts,
                           int* __restrict__ edges) {
  const int b = blockIdx.x;
  const int t = threadIdx.x;  // row j
  __shared__ int s0[256], s1[256];
  const float* Ab = A + (size_t)b * NN * NN;
  int c = 0;
  for (int k = t + 1; k < NN; ++k) c += (Ab[t * NN + k] != 0.0f) ? 1 : 0;
  s0[t] = c;
  __syncthreads();
  int* src = s0;
  int* dst = s1;
  for (int off = 1; off < 256; off <<= 1) {
    int v = src[t];
    if (t >= off) v += src[t - off];
    dst[t] = v;
    __syncthreads();
    int* tmp = src; src = dst; dst = tmp;
  }
  const int excl = src[t] - c;
  if (t == 0) {
    int total = src[255];
    cnts[b] = total < MAXE ? total : MAXE;
  }
  int e = excl;
  int* eb = edges + (size_t)b * MAXE * 2;
  for (int k = t + 1; k < NN; ++k) {
    if (Ab[t * NN + k] != 0.0f) {
      if (e < MAXE) { eb[e * 2 + 0] = t; eb[e * 2 + 1] = k; }
      ++e;
    }
  }
}

// ---------------------------------------------------------------------------
// K2: X_std = A @ X + (1+eps)*X. Block = 16-row panel of one batch; the A tile
// is async-staged to LDS once and shared by all 8 waves (one D-tile each).
// ---------------------------------------------------------------------------
__global__ void xstd_kernel(const float* __restrict__ A,
                            const float* __restrict__ X,
                            const float* __restrict__ epsp,
                            float* __restrict__ Xstd) {
  __shared__ float as[16][260];  // 260 % 64 == 4 -> conflict-free column reads
  const int b = blockIdx.y;
  const int mt = blockIdx.x;  // 16 row-panels
  const int tid = threadIdx.x;
  const float* Ab = A + (size_t)b * NN * NN;
  const float* Xb = X + (size_t)b * NN * DD;
  // stage A[mt*16 .. +16][0..256) -> LDS, 16B chunks (64 chunks per row)
  for (int c = tid; c < 16 * 64; c += 256) {
    const int r = c >> 6, cc = c & 63;
    const unsigned lds_off =
        (unsigned)(uintptr_t)&as[0][0] + (unsigned)((r * 260 + cc * 4) * 4);
    async_copy_b128(lds_off, Ab + (size_t)(mt * 16 + r) * NN + cc * 4);
  }
  wait_async();
  __syncthreads();
  const int wave = tid >> 5, lane = tid & 31;
  const int l16 = lane & 15, half = lane >> 4;
  const int n = wave * 16 + l16;  // 8 waves cover DD=128
  v8f acc = {};
  for (int k0 = 0; k0 < NN; k0 += 4) {
    const int ka = k0 + half * 2;
    v2f a, bv;
    a.x = as[l16][ka]; a.y = as[l16][ka + 1];
    bv.x = Xb[ka * DD + n];
    bv.y = Xb[(ka + 1) * DD + n];
    acc = wmma_step(a, bv, acc);
  }
  const float ep = 1.0f + epsp[0];
  float* Ob = Xstd + (size_t)b * NN * DD;
  for (int v = 0; v < 8; ++v) {
    const int row = mt * 16 + v + 8 * half;
    Ob[row * DD + n] = acc[v] + ep * Xb[row * DD + n];
  }
}

// ---------------------------------------------------------------------------
// K3: per-edge pair MLP: Fp[e] = relu(relu((X[j]+X[k])@W3+b3)@W4+b4)
// 16 edges per block; pair-sums and hidden tile staged in padded LDS.
// ---------------------------------------------------------------------------
__global__ void pair_mlp(const float* __restrict__ X,
                         const int* __restrict__ cnts,
                         const int* __restrict__ edges,
                         const float* __restrict__ W3, const float* __restrict__ b3,
                         const float* __restrict__ W4, const float* __restrict__ b4,
                         float* __restrict__ Fp) {
  const int b = blockIdx.y;
  const int cnt = min(cnts[b], MAXE);
  const int e0 = blockIdx.x * 16;
  if (e0 >= cnt) return;  // uniform per block
  __shared__ float xs[16][132];
  __shared__ float hs[16][260];
  const float* Xb = X + (size_t)b * NN * DD;
  const int* eb = edges + (size_t)b * MAXE * 2;
  const int tid = threadIdx.x;
  for (int idx = tid; idx < 16 * DD; idx += 256) {
    const int r = idx >> 7, d = idx & 127;
    const int e = e0 + r;
    float v = 0.0f;
    if (e < cnt) {
      const int j = eb[e * 2 + 0], k = eb[e * 2 + 1];
      v = Xb[j * DD + d] + Xb[k * DD + d];
    }
    xs[r][d] = v;
  }
  __syncthreads();
  const int wave = tid >> 5, lane = tid & 31;
  const int l16 = lane & 15, half = lane >> 4;
  // layer 1: [16 x 128] @ [128 x 256] + b3, relu -> hs
  for (int ct = wave; ct < HD / 16; ct += 8) {
    const int n0 = ct * 16;
    v8f acc = {};
    for (int k0 = 0; k0 < DD; k0 += 4) {
      const int ka = k0 + half * 2;
      v2f a, bv;
      a.x = xs[l16][ka]; a.y = xs[l16][ka + 1];
      bv.x = W3[ka * HD + n0 + l16];
      bv.y = W3[(ka + 1) * HD + n0 + l16];
      acc = wmma_step(a, bv, acc);
    }
    const float bias = b3[n0 + l16];
    for (int v = 0; v < 8; ++v) {
      const float h = acc[v] + bias;
      hs[v + 8 * half][n0 + l16] = h > 0.0f ? h : 0.0f;
    }
  }
  __syncthreads();
  // layer 2: [16 x 256] @ [256 x 128] + b4, relu -> Fp
  {
    const int n0 = wave * 16;  // DD/16 == 8 tiles == 8 waves
    v8f acc = {};
    for (int k0 = 0; k0 < HD; k0 += 4) {
      const int ka = k0 + half * 2;
      v2f a, bv;
      a.x = hs[l16][ka]; a.y = hs[l16][ka + 1];
      bv.x = W4[ka * DD + n0 + l16];
      bv.y = W4[(ka + 1) * DD + n0 + l16];
      acc = wmma_step(a, bv, acc);
    }
    const float bias = b4[n0 + l16];
    float* Fb = Fp + (size_t)b * MAXE * DD;
    for (int v = 0; v < 8; ++v) {
      const int e = e0 + v + 8 * half;
      float f = acc[v] + bias;
      f = f > 0.0f ? f : 0.0f;
      if (e < cnt) Fb[e * DD + n0 + l16] = f;
    }
  }
}

// ---------------------------------------------------------------------------
// K4: Xn = Xstd + W_adj @ Fp, where W_adj[i,e] = A[i,j_e]*A[i,k_e] built on the
// fly into LDS and contracted with WMMA over the edge (K) dimension.
// ---------------------------------------------------------------------------
__global__ void xpw_kernel(const float* __restrict__ A,
                           const int* __restrict__ cnts,
                           const int* __restrict__ edges,
                           const float* __restrict__ Fp,
                           float* __restrict__ Xn /* Xstd, updated in place */) {
  const int b = blockIdx.y;
  const int mt = blockIdx.x;
  const int cnt = min(cnts[b], MAXE);
  __shared__ float wc[16][68];  // 68 % 64 == 4
  const float* Ab = A + (size_t)b * NN * NN;
  const int* eb = edges + (size_t)b * MAXE * 2;
  const float* Fb = Fp + (size_t)b * MAXE * DD;
  float* Ob = Xn + (size_t)b * NN * DD;
  const int tid = threadIdx.x;
  const int wave = tid >> 5, lane = tid & 31;
  const int l16 = lane & 15, half = lane >> 4;
  const int n = wave * 16 + l16;  // 8 waves cover DD=128
  v8f acc;
  for (int v = 0; v < 8; ++v) acc[v] = Ob[(mt * 16 + v + 8 * half) * DD + n];
  for (int c0 = 0; c0 < cnt; c0 += 64) {
    __syncthreads();
    for (int idx = tid; idx < 16 * 64; idx += 256) {
      const int r = idx >> 6, el = idx & 63;
      const int e = c0 + el;
      float w = 0.0f;
      if (e < cnt) {
        const int j = eb[e * 2 + 0], k = eb[e * 2 + 1];
        const int i = mt * 16 + r;
        w = Ab[i * NN + j] * Ab[i * NN + k];
      }
      wc[r][el] = w;
    }
    __syncthreads();
    for (int kk = 0; kk < 64; kk += 4) {
      const int ka = kk + half * 2;
      v2f a, bv;
      a.x = wc[l16][ka]; a.y = wc[l16][ka + 1];
      const int e = c0 + ka;
      bv.x = Fb[e * DD + n];
      bv.y = Fb[(e + 1) * DD + n];
      acc = wmma_step(a, bv, acc);
    }
  }
  for (int v = 0; v < 8; ++v) Ob[(mt * 16 + v + 8 * half) * DD + n] = acc[v];
}

// ---------------------------------------------------------------------------
// K5/K7: Out = relu(In @ W + bias). Block = 16-row panel; the In tile is
// async-staged to LDS once (ASYNCcnt) and reused by all waves/column-tiles.
// ---------------------------------------------------------------------------
__global__ void gemm_bias_relu(const float* __restrict__ In,
                               const float* __restrict__ W,
                               const float* __restrict__ bias,
                               float* __restrict__ Out, int K, int Ncols) {
  __shared__ float xs[16][260];  // sized for K<=256; row stride = K+4
  const int mt = blockIdx.x;
  const int tid = threadIdx.x;
  const int pad = K + 4;  // (K+4) % 64 == 4 for K in {128,256}
  float* xsf = &xs[0][0];
  const int chunksPerRow = K >> 2;  // 16B chunks per row
  for (int c = tid; c < 16 * chunksPerRow; c += 256) {
    const int r = c / chunksPerRow, cc = c % chunksPerRow;
    const unsigned lds_off =
        (unsigned)(uintptr_t)xsf + (unsigned)((r * pad + cc * 4) * 4);
    async_copy_b128(lds_off, In + (size_t)(mt * 16 + r) * K + cc * 4);
  }
  wait_async();
  __syncthreads();
  const int wave = tid >> 5, lane = tid & 31;
  const int l16 = lane & 15, half = lane >> 4;
  for (int nt = wave; nt < Ncols / 16; nt += 8) {
    const int n = nt * 16 + l16;
    v8f acc = {};
    for (int k0 = 0; k0 < K; k0 += 4) {
      const int ka = k0 + half * 2;
      v2f a, bv;
      a.x = xsf[l16 * pad + ka];
      a.y = xsf[l16 * pad + ka + 1];
      bv.x = W[ka * Ncols + n];
      bv.y = W[(ka + 1) * Ncols + n];
      acc = wmma_step(a, bv, acc);
    }
    const float bs = bias[n];
    for (int v = 0; v < 8; ++v) {
      const float h = acc[v] + bs;
      Out[(size_t)(mt * 16 + v + 8 * half) * Ncols + n] = h > 0.0f ? h : 0.0f;
    }
  }
}

// ---------------------------------------------------------------------------
// K6/K8: BatchNorm1d (training stats, biased var) over channel n: stats across
// (batch, last dim) = 1024 values. One block per channel, tree reduction.
// ---------------------------------------------------------------------------
__global__ void bn_kernel(const float* __restrict__ In,
                          const float* __restrict__ gamma,
                          const float* __restrict__ beta,
                          float* __restrict__ Out) {
  const int n = blockIdx.x;   // channel 0..255
  const int h = threadIdx.x;  // 0..255
  __shared__ float rs[256], rq[256];
  float vals[BB];
  float s = 0.0f, q = 0.0f;
  for (int b = 0; b < BB; ++b) {
    const float v = In[((size_t)(b * NN + n)) * HD + h];
    vals[b] = v;
    s += v; q += v * v;
  }
  rs[h] = s; rq[h] = q;
  __syncthreads();
  for (int off = 128; off > 0; off >>= 1) {
    if (h < off) { rs[h] += rs[h + off]; rq[h] += rq[h + off]; }
    __syncthreads();
  }
  const float inv_cnt = 1.0f / (float)(BB * HD);
  const float mean = rs[0] * inv_cnt;
  const float var = rq[0] * inv_cnt - mean * mean;
  const float inv = rsqrtf(var + BN_EPS);
  const float gm = gamma[n], bt = beta[n];
  for (int b = 0; b < BB; ++b)
    Out[((size_t)(b * NN + n)) * HD + h] = gm * (vals[b] - mean) * inv + bt;
}

extern "C" void kernel_launch(void* const* d_in, const int* in_sizes, int n_in,
                              void* d_out, int out_size, void* d_ws,
                              size_t ws_size, hipStream_t stream) {
  const float* A   = (const float*)d_in[0];
  const float* X   = (const float*)d_in[1];
  const float* eps = (const float*)d_in[2];
  const float* W1  = (const float*)d_in[3];
  const float* b1  = (const float*)d_in[4];
  const float* g1  = (const float*)d_in[5];
  const float* be1 = (const float*)d_in[6];
  const float* W2  = (const float*)d_in[7];
  const float* b2  = (const float*)d_in[8];
  const float* g2  = (const float*)d_in[9];
  const float* be2 = (const float*)d_in[10];
  const float* W3  = (const float*)d_in[11];
  const float* b3  = (const float*)d_in[12];
  const float* W4  = (const float*)d_in[13];
  const float* b4  = (const float*)d_in[14];

  char* ws = (char*)d_ws;
  int* cnts  = (int*)ws;                               // 64 B
  int* edges = (int*)(ws + 64);                        // B*MAXE*2 ints
  size_t off = 64 + (size_t)BB * MAXE * 2 * sizeof(int);
  float* Xstd = (float*)(ws + off); off += (size_t)BB * NN * DD * sizeof(float);
  float* Fp   = (float*)(ws + off); off += (size_t)BB * MAXE * DD * sizeof(float);
  float* Y1   = (float*)(ws + off); off += (size_t)BB * NN * HD * sizeof(float);
  float* h1   = (float*)(ws + off); off += (size_t)BB * NN * HD * sizeof(float);
  float* Y2   = (float*)(ws + off); off += (size_t)BB * NN * HD * sizeof(float);
  float* out  = (float*)d_out;

  edge_build<<<BB, 256, 0, stream>>>(A, cnts, edges);
  xstd_kernel<<<dim3(NN / 16, BB), 256, 0, stream>>>(A, X, eps, Xstd);
  pair_mlp<<<dim3(MAXE / 16, BB), 256, 0, stream>>>(X, cnts, edges, W3, b3, W4,
                                                    b4, Fp);
  xpw_kernel<<<dim3(NN / 16, BB), 256, 0, stream>>>(A, cnts, edges, Fp, Xstd);
  gemm_bias_relu<<<BB * NN / 16, 256, 0, stream>>>(Xstd, W1, b1, Y1, DD, HD);
  bn_kernel<<<NN, 256, 0, stream>>>(Y1, g1, be1, h1);
  gemm_bias_relu<<<BB * NN / 16, 256, 0, stream>>>(h1, W2, b2, Y2, HD, HD);
  bn_kernel<<<NN, 256, 0, stream>>>(Y2, g2, be2, out);
}